// NeRF_52707838656839
// MI455X (gfx1250) — compile-verified
//
#include <hip/hip_runtime.h>

// ---------------------------------------------------------------------------
// NeRF volume rendering, MI455X (gfx1250).
// Memory-bound: ~819 MB in / 23.3 TB/s ~= 35 us floor. Strategy: stream
// sigma/t/rgb through LDS with CDNA5 async global->LDS b128 copies,
// double-buffered, overlapping the per-pixel transmittance scan with the
// next chunk's HBM fetch. One thread = one pixel (sequential cumprod along S).
// ---------------------------------------------------------------------------

#ifndef __has_builtin
#define __has_builtin(x) 0
#endif

constexpr int H    = 800;
constexpr int W    = 800;
constexpr int S    = 64;
constexpr int NPIX = H * W;

constexpr int PPB = 128;      // pixels per block (4 waves of 32)
constexpr int CS  = 8;        // samples per chunk
constexpr int NC  = S / CS;   // 8 chunks
// async issues per lane per chunk: 2 (sigma) + 2 (t) + 6 (rgb) = 10
constexpr int ISSUES_PER_CHUNK = 10;

// Async DMA: global (GV mode, 64-bit vaddr) -> LDS, 16 bytes per lane.
// vdst = VGPR holding wave-relative LDS byte address; generic pointers to
// __shared__ objects carry that address in their low 32 bits (ISA 10.2).
__device__ __forceinline__ void async_gld_b128(const void* g, void* l) {
  unsigned lofs = (unsigned)(unsigned long long)l;      // LDS byte address
  unsigned long long ga = (unsigned long long)g;        // global address
  asm volatile("global_load_async_to_lds_b128 %0, %1, off"
               :: "v"(lofs), "v"(ga) : "memory");
}

template <int N>
__device__ __forceinline__ void wait_asynccnt() {
#if __has_builtin(__builtin_amdgcn_s_wait_asynccnt)
  __builtin_amdgcn_s_wait_asynccnt(N);
#else
  asm volatile("s_wait_asynccnt %0" :: "i"(N) : "memory");
#endif
}

__global__ __launch_bounds__(PPB) void nerf_render_kernel(
    const float* __restrict__ rgb,     // [H,W,S,3]
    const float* __restrict__ sig,     // [H,W,S]
    const float* __restrict__ tt,      // [H,W,S]
    float* __restrict__ out)           // [H,W,3] ++ [H,W]
{
  __shared__ __align__(16) float ls_sig[2][PPB * CS];        //  8 KB
  __shared__ __align__(16) float ls_t  [2][PPB * CS];        //  8 KB
  __shared__ __align__(16) float ls_rgb[2][PPB * CS * 3];    // 24 KB  (x2 bufs)

  const int tid = threadIdx.x;
  const int p0  = blockIdx.x * PPB;
  const int p   = p0 + tid;

  // Cooperative async staging of chunk c into LDS buffer b.
  // Each lane moves 10x16B. All addresses 16B aligned (CS*4B = 32B runs).
  auto issue = [&](int c, int b) {
    const int c0 = c * CS;
#pragma unroll
    for (int k = 0; k < 2; ++k) {                 // sigma + t: 256 segs each
      int seg  = tid + k * PPB;                   // 0..255
      int pix  = seg >> 1;
      int half = seg & 1;
      size_t gof = (size_t)(p0 + pix) * S + c0 + half * 4;   // float index
      int    lof = pix * CS + half * 4;
      async_gld_b128(sig + gof, &ls_sig[b][lof]);
      async_gld_b128(tt  + gof, &ls_t  [b][lof]);
    }
#pragma unroll
    for (int k = 0; k < 6; ++k) {                 // rgb: 768 segs
      int seg = tid + k * PPB;                    // 0..767
      int pix = seg / 6;
      int off = seg - pix * 6;
      size_t gof = (size_t)(p0 + pix) * (S * 3) + c0 * 3 + off * 4;
      async_gld_b128(rgb + gof, &ls_rgb[b][pix * (CS * 3) + off * 4]);
    }
  };

  float T  = 1.0f;                       // exclusive transmittance (cumprod)
  float aR = 0.f, aG = 0.f, aB = 0.f, aD = 0.f;
  float pt = 0.f, ps = 0.f, pr = 0.f, pg = 0.f, pb = 0.f;  // carried sample

  issue(0, 0);

  for (int c = 0; c < NC; ++c) {
    const int b = c & 1;
    if (c + 1 < NC) {
      issue(c + 1, b ^ 1);               // prefetch next chunk (other buffer)
      wait_asynccnt<ISSUES_PER_CHUNK>(); // in-order completion => chunk c done
    } else {
      wait_asynccnt<0>();
    }
    __syncthreads();                     // everyone's chunk-c data visible

    const float* cs = &ls_sig[b][tid * CS];
    const float* ct = &ls_t  [b][tid * CS];
    const float* cr = &ls_rgb[b][tid * CS * 3];

#pragma unroll
    for (int s = 0; s < CS; ++s) {
      float ts = ct[s];
      float sg = cs[s];
      float r  = cr[s * 3 + 0];
      float g  = cr[s * 3 + 1];
      float bl = cr[s * 3 + 2];
      if (c | s) {                       // global sample index > 0
        float delta = ts - pt;           // t[i+1] - t[i]
        float sd    = __expf(-ps * delta);
        float w     = (1.0f - sd) * T;   // alpha * exclusive transmittance
        aR += w * pr; aG += w * pg; aB += w * pb; aD += w * pt;
        T  *= sd + 1e-9f;                // cumprod(sig_dels + EPS)
      }
      pt = ts; ps = sg; pr = r; pg = g; pb = bl;
    }
    __syncthreads();                     // buffer b reusable by issue(c+2)
  }

  // Final sample: delta = INF sentinel (1e9), matches reference.
  {
    float sd = __expf(ps * -1.0e9f);
    float w  = (1.0f - sd) * T;
    aR += w * pr; aG += w * pg; aB += w * pb; aD += w * pt;
  }

  out[p * 3 + 0]    = fminf(aR, 1.0f);
  out[p * 3 + 1]    = fminf(aG, 1.0f);
  out[p * 3 + 2]    = fminf(aB, 1.0f);
  out[NPIX * 3 + p] = aD;
}

extern "C" void kernel_launch(void* const* d_in, const int* in_sizes, int n_in,
                              void* d_out, int out_size, void* d_ws, size_t ws_size,
                              hipStream_t stream) {
  (void)in_sizes; (void)n_in; (void)out_size; (void)d_ws; (void)ws_size;
  const float* rgb = (const float*)d_in[0];   // [H,W,S,3]
  const float* sig = (const float*)d_in[1];   // [H,W,S]
  const float* t   = (const float*)d_in[2];   // [H,W,S]
  float* out = (float*)d_out;                 // [H,W,3] ++ [H,W]

  dim3 grid(NPIX / PPB);                      // 5000 blocks, exact cover
  dim3 block(PPB);
  nerf_render_kernel<<<grid, block, 0, stream>>>(rgb, sig, t, out);
}